// ProbaVLoss_45440753991929
// MI455X (gfx1250) — compile-verified
//
#include <hip/hip_runtime.h>

// ---------------------------------------------------------------------------
// ProbaV loss for MI455X (gfx1250, wave32).
//   predict/target/mask: (8,1,384,384) f32.  Output: scalar f32.
//   36 shifts x 8 images x 378x378 pixels; per pixel: zero-padded 3x3 means
//   of {x, y, x^2, y^2, xy} -> SSIM, plus diff stats for bias-corrected MSE.
// One wave32 handles one 16x16 output tile (lane = column x half-row-group,
// matching the CDNA5 WMMA C/D layout). Cross-lane reduction: chained
// V_WMMA_F32_16X16X4_F32 (full fp32 matrix op, co-executes with VALU) for the
// half-lane pair sums, then an immediate DS_SWIZZLE xor-butterfly (no address
// VGPRs / bounds checks, unlike ds_bpermute) for the 16-column sum.
// ---------------------------------------------------------------------------

#define IMG_H 384
#define IMG_W 384
#define CROPPED 378           // 384 - 2*3
#define N_SHIFT 36
#define N_BATCH 8
#define TILES_PER_DIM 24      // ceil(378/16)
#define TILES_PER_IMG (TILES_PER_DIM * TILES_PER_DIM)   // 576
#define WAVES_PER_BLOCK 8
#define BLOCKS_PER_IMG (TILES_PER_IMG / WAVES_PER_BLOCK) // 72
#define TOTAL_BLOCKS (N_SHIFT * N_BATCH * BLOCKS_PER_IMG) // 20736
#define LDS_STRIDE 19         // 18 halo cols padded to odd stride (bank friendly)

typedef float v2f __attribute__((ext_vector_type(2)));
typedef float v8f __attribute__((ext_vector_type(8)));

// DS_SWIZZLE_B32 group-of-32 mode: offset[4:0]=and_mask, [9:5]=or_mask,
// [14:10]=xor_mask; new_lane = ((lane & and) | or) ^ xor.
#define SWZ_XOR1 0x041F
#define SWZ_XOR2 0x081F
#define SWZ_XOR4 0x101F
#define SWZ_XOR8 0x201F
#define SWZ_ADD(v, imm) \
    ((v) += __int_as_float(__builtin_amdgcn_ds_swizzle(__float_as_int(v), (imm))))

__global__ void __launch_bounds__(256)
probav_tile_kernel(const float* __restrict__ predict,
                   const float* __restrict__ target,
                   const float* __restrict__ mask,
                   float* __restrict__ partials) {
    const int bid = blockIdx.x;                // [0, 36*8*72)
    const int tb  = bid % BLOCKS_PER_IMG;      // tile-block within image
    const int sb  = bid / BLOCKS_PER_IMG;      // s*8 + b
    const int b   = sb & 7;
    const int s   = sb >> 3;
    const int u   = s / 6;                     // row shift 0..5
    const int v   = s % 6;                     // col shift 0..5

    const int lane = threadIdx.x & 31;
    const int wave = threadIdx.x >> 5;
    const int tile = tb * WAVES_PER_BLOCK + wave;       // 0..575
    const int ti   = tile / TILES_PER_DIM;
    const int tj   = tile % TILES_PER_DIM;
    const int i0   = ti * 16;
    const int j0   = tj * 16;

    // Wave-private halo tiles (18x18 region, stride 19): x = masked crop,
    // y = masked center-crop target. 2 * 8 * 342 * 4B = 21.9 KB of 320KB LDS.
    __shared__ float ldsx[WAVES_PER_BLOCK][18 * LDS_STRIDE];
    __shared__ float ldsy[WAVES_PER_BLOCK][18 * LDS_STRIDE];

    const float* pB = predict + (size_t)b * (IMG_H * IMG_W);
    const float* tB = target  + (size_t)b * (IMG_H * IMG_W);
    const float* mB = mask    + (size_t)b * (IMG_H * IMG_W);

    // ---- fill 18x18 halo (zero-padded at the 378x378 crop boundary) ----
    for (int idx = lane; idx < 324; idx += 32) {
        const int hr = idx / 18;
        const int hc = idx - hr * 18;
        const int ic = i0 - 1 + hr;            // crop-space row
        const int jc = j0 - 1 + hc;            // crop-space col
        float xv = 0.0f, yv = 0.0f;
        if (ic >= 0 && ic < CROPPED && jc >= 0 && jc < CROPPED) {
            const int pr = u + ic, pc = v + jc;        // predict coords
            const int tr = 3 + ic, tc = 3 + jc;        // target coords
            xv = pB[pr * IMG_W + pc] * mB[pr * IMG_W + pc];
            yv = tB[tr * IMG_W + tc] * mB[tr * IMG_W + tc];
        }
        ldsx[wave][hr * LDS_STRIDE + hc] = xv;
        ldsy[wave][hr * LDS_STRIDE + hc] = yv;
    }
    __syncthreads();

    // ---- per-lane streaming: one column, 8 rows (C/D layout mapping) ----
    const int n   = lane & 15;                 // tile column
    const int m0  = (lane >> 4) * 8;           // first tile row for this lane
    const int j   = j0 + n;                    // crop-space column
    const bool jok = (j < CROPPED);

    const float* lx = ldsx[wave];
    const float* ly = ldsy[wave];

    float hx[3], hy[3], hxx[3], hyy[3], hxy[3], cx[3], cy[3];
    float p_ds = 0.0f, p_d = 0.0f, p_d2 = 0.0f;

    const float inv9 = 1.0f / 9.0f;
    const float K1 = 1.0e-4f;                  // 0.01^2
    const float K2 = 9.0e-4f;                  // 0.03^2

#pragma unroll
    for (int rr = 0; rr < 10; ++rr) {
        const int hr   = m0 + rr;                        // halo row index
        const int base = hr * LDS_STRIDE + n;            // halo col n == crop col j-1
        const float x0 = lx[base], x1 = lx[base + 1], x2 = lx[base + 2];
        const float y0 = ly[base], y1 = ly[base + 1], y2 = ly[base + 2];
        const int k = rr % 3;
        hx[k]  = x0 + x1 + x2;
        hy[k]  = y0 + y1 + y2;
        hxx[k] = x0 * x0 + x1 * x1 + x2 * x2;
        hyy[k] = y0 * y0 + y1 * y1 + y2 * y2;
        hxy[k] = x0 * y0 + x1 * y1 + x2 * y2;
        cx[k]  = x1;
        cy[k]  = y1;
        if (rr >= 2) {
            const int r = rr - 2;                        // output row in [0,8)
            const float Sx  = hx[0]  + hx[1]  + hx[2];
            const float Sy  = hy[0]  + hy[1]  + hy[2];
            const float Sxx = hxx[0] + hxx[1] + hxx[2];
            const float Syy = hyy[0] + hyy[1] + hyy[2];
            const float Sxy = hxy[0] + hxy[1] + hxy[2];
            const int km = (rr - 1) % 3;                 // center row of window
            const float xc = cx[km], yc = cy[km];

            const float mux = Sx * inv9, muy = Sy * inv9;
            const float sxx = Sxx * inv9 - mux * mux;
            const float syy = Syy * inv9 - muy * muy;
            const float sxy = Sxy * inv9 - mux * muy;
            const float num = (2.0f * mux * muy + K1) * (2.0f * sxy + K2);
            const float den = (mux * mux + muy * muy + K1) * (sxx + syy + K2);
            const float ssim = num / den;
            float ds = (1.0f - ssim) * 0.5f;
            ds = fminf(fmaxf(ds, 0.0f), 1.0f);

            const float diff = yc - xc;                  // tgt - crop
            const bool ok = jok && ((i0 + m0 + r) < CROPPED);
            const float w = ok ? 1.0f : 0.0f;
            p_ds += w * ds;
            p_d  += w * diff;
            p_d2 += w * diff * diff;
        }
    }

    // ---- cross-lane reduction on the matrix pipe (fp32 WMMA) ----
    // A[q][k] = 1 for all k (single-hot row q), B = (partial, 0).
    // Since the selected A row is all-ones across K, D[q][n] = p(n) + p(16+n)
    // regardless of the K<->VGPR interleave -> layout-robust exact f32 sums.
    // EXEC is all 1s here (no early returns), as WMMA requires.
    const int msel = lane & 15;
    v2f a0; a0.x = a0.y = (msel == 0) ? 1.0f : 0.0f;
    v2f a1; a1.x = a1.y = (msel == 1) ? 1.0f : 0.0f;
    v2f a2; a2.x = a2.y = (msel == 2) ? 1.0f : 0.0f;
    v2f b0; b0.x = p_ds; b0.y = 0.0f;
    v2f b1; b1.x = p_d;  b1.y = 0.0f;
    v2f b2; b2.x = p_d2; b2.y = 0.0f;
    v8f acc = {0.f, 0.f, 0.f, 0.f, 0.f, 0.f, 0.f, 0.f};
    acc = __builtin_amdgcn_wmma_f32_16x16x4_f32(false, a0, false, b0, (short)0, acc, false, false);
    acc = __builtin_amdgcn_wmma_f32_16x16x4_f32(false, a1, false, b1, (short)0, acc, false, false);
    acc = __builtin_amdgcn_wmma_f32_16x16x4_f32(false, a2, false, b2, (short)0, acc, false, false);

    // Lanes 0-15 hold rows 0..2 (ds, d, d2 column-pair sums) in acc[0..2];
    // lanes 16-31 hold zero rows. Finish the 16-column sum with an immediate
    // DS_SWIZZLE xor-butterfly (upper half adds zeros among itself).
    float r_ds = acc[0];
    float r_d  = acc[1];
    float r_d2 = acc[2];
    SWZ_ADD(r_ds, SWZ_XOR1); SWZ_ADD(r_d, SWZ_XOR1); SWZ_ADD(r_d2, SWZ_XOR1);
    SWZ_ADD(r_ds, SWZ_XOR2); SWZ_ADD(r_d, SWZ_XOR2); SWZ_ADD(r_d2, SWZ_XOR2);
    SWZ_ADD(r_ds, SWZ_XOR4); SWZ_ADD(r_d, SWZ_XOR4); SWZ_ADD(r_d2, SWZ_XOR4);
    SWZ_ADD(r_ds, SWZ_XOR8); SWZ_ADD(r_d, SWZ_XOR8); SWZ_ADD(r_d2, SWZ_XOR8);

    // ---- deterministic per-block combine (no float atomics) ----
    __shared__ float red[WAVES_PER_BLOCK][3];
    if (lane == 0) {
        red[wave][0] = r_ds;
        red[wave][1] = r_d;
        red[wave][2] = r_d2;
    }
    __syncthreads();
    if (threadIdx.x < 3) {
        float t = 0.0f;
#pragma unroll
        for (int wv = 0; wv < WAVES_PER_BLOCK; ++wv) t += red[wv][threadIdx.x];
        partials[(size_t)bid * 3 + threadIdx.x] = t;
    }
}

// One block per shift: fixed-order reduction of its 576 block-partials, then
// loss[s] = (E[d^2] - (E[d])^2) + 5 * E[ds].
__global__ void __launch_bounds__(256)
probav_shift_reduce(const float* __restrict__ partials, float* __restrict__ loss) {
    const int s = blockIdx.x;
    const int t = threadIdx.x;
    __shared__ float sm[256][3];
    float a0 = 0.0f, a1 = 0.0f, a2 = 0.0f;
    for (int k = t; k < N_BATCH * BLOCKS_PER_IMG; k += 256) {
        const float* p = partials + (size_t)(s * (N_BATCH * BLOCKS_PER_IMG) + k) * 3;
        a0 += p[0]; a1 += p[1]; a2 += p[2];
    }
    sm[t][0] = a0; sm[t][1] = a1; sm[t][2] = a2;
    __syncthreads();
    for (int off = 128; off > 0; off >>= 1) {
        if (t < off) {
            sm[t][0] += sm[t + off][0];
            sm[t][1] += sm[t + off][1];
            sm[t][2] += sm[t + off][2];
        }
        __syncthreads();
    }
    if (t == 0) {
        const float invN = 1.0f / (8.0f * 378.0f * 378.0f);
        const float sum_ds = sm[0][0];
        const float sum_d  = sm[0][1];
        const float sum_d2 = sm[0][2];
        const float bb = sum_d * invN;
        const float mse = sum_d2 * invN - bb * bb;
        loss[s] = mse + 5.0f * (sum_ds * invN);
    }
}

__global__ void __launch_bounds__(64)
probav_min_kernel(const float* __restrict__ loss, float* __restrict__ out) {
    const int t = threadIdx.x;
    __shared__ float sm[64];
    sm[t] = (t < N_SHIFT) ? loss[t] : 3.4e38f;
    __syncthreads();
    for (int off = 32; off > 0; off >>= 1) {
        if (t < off) sm[t] = fminf(sm[t], sm[t + off]);
        __syncthreads();
    }
    if (t == 0) out[0] = sm[0];
}

extern "C" void kernel_launch(void* const* d_in, const int* in_sizes, int n_in,
                              void* d_out, int out_size, void* d_ws, size_t ws_size,
                              hipStream_t stream) {
    const float* predict = (const float*)d_in[0];
    const float* target  = (const float*)d_in[1];
    const float* mask    = (const float*)d_in[2];
    float* out = (float*)d_out;

    float* partials = (float*)d_ws;                        // 20736 * 3 floats (~243 KB)
    float* loss     = partials + (size_t)TOTAL_BLOCKS * 3; // 36 floats

    probav_tile_kernel<<<TOTAL_BLOCKS, 256, 0, stream>>>(predict, target, mask, partials);
    probav_shift_reduce<<<N_SHIFT, 256, 0, stream>>>(partials, loss);
    probav_min_kernel<<<1, 64, 0, stream>>>(loss, out);
}